// Sparse2L_41807211659989
// MI455X (gfx1250) — compile-verified
//
#include <hip/hip_runtime.h>

typedef __attribute__((ext_vector_type(2))) float v2f;
typedef __attribute__((ext_vector_type(8))) float v8f;

#define GAMMA 0.1f
#define RHO   0.9f
#define LR    0.009f
#define MOM   0.9f
#define EPSR  1e-8f

#define B_DIM 2048
#define DIN   1024
#define DOUT  2048
#define DC    256
#define NSTEP 10

#define BLK_M 128
#define BLK_N 64
#define KC    32
#define LPAD  4          // row stride KC+4: 16B-aligned rows, conflict-free

struct WaveAcc {
  v8f c[2][2];   // [mi][nj] 16x16 tiles; wave covers 32x32
  int rbase;     // global row of c[i][*] element r -> rbase + i*16 + r
  int cbase;     // global col of c[*][j]           -> cbase + j*16
};

// Block-level fp32 GEMM core: C_tile(128x64) += A(128xK) * B(Kx64).
// 8 wave32 waves (4M x 2N); each wave register-blocks a 2x2 grid of 16x16
// WMMA tiles. Both A and B tiles are staged K-contiguous per row so every
// fragment is one aligned ds_load_b64 (no register shuffling before wmma).
// TRANSB=false: Bm row-major [K x N], row stride ldb.
// TRANSB=true : Bm row-major [N x K], row stride ldb (B[k][n] = Bm[n][k]).
template <bool TRANSB>
__device__ __forceinline__ WaveAcc gemm_core(const float* __restrict__ A, int lda,
                                             const float* __restrict__ Bm, int ldb,
                                             int K, int m0, int n0) {
  __shared__ float sA[BLK_M][KC + LPAD];   // [m][k]
  __shared__ float sB[BLK_N][KC + LPAD];   // [n][k]  (n-major!)

  const int t    = threadIdx.x;
  const int lane = t & 31;
  const int wave = t >> 5;
  const int hf   = lane >> 4;   // 0: K pair {0,1}, 1: K pair {2,3}
  const int lid  = lane & 15;
  const int wr   = wave & 3;    // wave row (4 along M, 32 rows each)
  const int wc   = wave >> 2;   // wave col (2 along N, 32 cols each)

  // cooperative-load geometry
  const int arow = t >> 1;        // 0..127
  const int acol = (t & 1) * 16;  // 0 or 16 -> four float4s
  const int brow = t >> 3;        // 0..31  (k row)     [!TRANSB]
  const int bcol = (t & 7) * 8;   // 0..56  -> two float4s along N
  const int tn   = t >> 2;        // 0..63  (n row)     [TRANSB]
  const int tk   = (t & 3) * 8;   // 0..24  -> two float4s along K

  WaveAcc w;
#pragma unroll
  for (int i = 0; i < 2; ++i)
#pragma unroll
    for (int j = 0; j < 2; ++j)
      w.c[i][j] = (v8f){0.f, 0.f, 0.f, 0.f, 0.f, 0.f, 0.f, 0.f};

  for (int k0 = 0; k0 < K; k0 += KC) {
    // stage A tile (128 x 32): 16 floats/thread, float4 global -> float4 LDS
    {
      const float* ap = A + (size_t)(m0 + arow) * lda + (k0 + acol);
#pragma unroll
      for (int q = 0; q < 4; ++q)
        *(float4*)&sA[arow][acol + 4 * q] = *(const float4*)(ap + 4 * q);
    }

    // stage B tile as sB[n][k]
    if (!TRANSB) {
      // Bm[K x N]: read float4 along N (coalesced), scatter-transpose into LDS
      const float* bp = Bm + (size_t)(k0 + brow) * ldb + (n0 + bcol);
#pragma unroll
      for (int q = 0; q < 2; ++q) {
        const float4 f = *(const float4*)(bp + 4 * q);
        sB[bcol + 4 * q + 0][brow] = f.x;
        sB[bcol + 4 * q + 1][brow] = f.y;
        sB[bcol + 4 * q + 2][brow] = f.z;
        sB[bcol + 4 * q + 3][brow] = f.w;
      }
    } else {
      // Bm[N x K]: read float4 along K (coalesced), store contiguous rows
      const float* bp = Bm + (size_t)(n0 + tn) * ldb + (k0 + tk);
#pragma unroll
      for (int q = 0; q < 2; ++q)
        *(float4*)&sB[tn][tk + 4 * q] = *(const float4*)(bp + 4 * q);
    }

    // hint-prefetch next K tile of A into GL2 (global_prefetch_b8)
    if (k0 + KC < K)
      __builtin_prefetch(A + (size_t)(m0 + arow) * lda + (k0 + KC + acol), 0, 0);

    __syncthreads();

#pragma unroll
    for (int kk = 0; kk < KC; kk += 4) {
      const int kb  = kk + 2 * hf;          // even -> 8B-aligned LDS pairs
      const int ar0 = wr * 32 + lid;
      const int bc0 = wc * 32 + lid;
      const v2f a0 = *(const v2f*)&sA[ar0][kb];
      const v2f a1 = *(const v2f*)&sA[ar0 + 16][kb];
      const v2f b0 = *(const v2f*)&sB[bc0][kb];
      const v2f b1 = *(const v2f*)&sB[bc0 + 16][kb];
      w.c[0][0] = __builtin_amdgcn_wmma_f32_16x16x4_f32(false, a0, false, b0,
                                                        (short)0, w.c[0][0], false, false);
      w.c[0][1] = __builtin_amdgcn_wmma_f32_16x16x4_f32(false, a0, false, b1,
                                                        (short)0, w.c[0][1], false, false);
      w.c[1][0] = __builtin_amdgcn_wmma_f32_16x16x4_f32(false, a1, false, b0,
                                                        (short)0, w.c[1][0], false, false);
      w.c[1][1] = __builtin_amdgcn_wmma_f32_16x16x4_f32(false, a1, false, b1,
                                                        (short)0, w.c[1][1], false, false);
    }
    __syncthreads();
  }

  w.rbase = m0 + wr * 32 + 8 * hf;
  w.cbase = n0 + wc * 32 + lid;
  return w;
}

// ---- step kernel 1: E = exp(-(u @ V))   [B x DOUT] ----
__global__ __launch_bounds__(256) void k1_uV_exp(const float* __restrict__ u,
                                                 const float* __restrict__ V,
                                                 float* __restrict__ E) {
  WaveAcc w = gemm_core<false>(u, DC, V, DOUT, DC,
                               blockIdx.x * BLK_M, blockIdx.y * BLK_N);
#pragma unroll
  for (int i = 0; i < 2; ++i)
#pragma unroll
    for (int j = 0; j < 2; ++j)
#pragma unroll
      for (int r = 0; r < 8; ++r)
        E[(size_t)(w.rbase + i * 16 + r) * DOUT + (w.cbase + j * 16)] =
            __expf(-w.c[i][j][r]);
}

// ---- step kernel 2: R = x @ W - inputs   [B x DIN] ----
__global__ __launch_bounds__(256) void k2_xW_res(const float* __restrict__ x,
                                                 const float* __restrict__ W,
                                                 const float* __restrict__ inp,
                                                 float* __restrict__ R) {
  WaveAcc w = gemm_core<false>(x, DOUT, W, DIN, DOUT,
                               blockIdx.x * BLK_M, blockIdx.y * BLK_N);
#pragma unroll
  for (int i = 0; i < 2; ++i)
#pragma unroll
    for (int j = 0; j < 2; ++j)
#pragma unroll
      for (int r = 0; r < 8; ++r) {
        const size_t o = (size_t)(w.rbase + i * 16 + r) * DIN + (w.cbase + j * 16);
        R[o] = w.c[i][j][r] - inp[o];
      }
}

// ---- step kernel 3: gx = 2 R@W^T + (G*causes + G/10)*x/dabs(x);
//      P = -0.5*G*dabs(x)*E (in place over E); rmsprop update of x,a1,a2 ----
__global__ __launch_bounds__(256) void k3_gx_update(const float* __restrict__ Rm,
                                                    const float* __restrict__ W,
                                                    float* __restrict__ x,
                                                    float* __restrict__ a1,
                                                    float* __restrict__ a2,
                                                    float* __restrict__ EP) {
  WaveAcc w = gemm_core<true>(Rm, DIN, W, DIN, DIN,
                              blockIdx.x * BLK_M, blockIdx.y * BLK_N);
#pragma unroll
  for (int i = 0; i < 2; ++i)
#pragma unroll
    for (int j = 0; j < 2; ++j)
#pragma unroll
      for (int r = 0; r < 8; ++r) {
        const size_t o = (size_t)(w.rbase + i * 16 + r) * DOUT + (w.cbase + j * 16);
        const float xv  = x[o];
        const float Ev  = EP[o];
        const float dab = sqrtf(xv * xv + 1e-6f);
        const float g   = 2.f * w.c[i][j][r] +
                          (GAMMA * (0.5f * (1.f + Ev)) + GAMMA * 0.1f) * (xv / dab);
        EP[o] = GAMMA * dab * (-0.5f * Ev);      // P for gu GEMM (uses old x)
        const float a1n = RHO * a1[o] + (1.f - RHO) * g * g;
        const float upd = LR * g / sqrtf(a1n + EPSR);
        const float a2n = MOM * a2[o] - upd;
        a1[o] = a1n;
        a2[o] = a2n;
        x[o]  = xv + MOM * a2n - upd;
      }
}

// ---- step kernel 4: gu = P @ V^T; rmsprop update of u,b1,b2   [B x DC] ----
__global__ __launch_bounds__(256) void k4_gu_update(const float* __restrict__ P,
                                                    const float* __restrict__ V,
                                                    float* __restrict__ u,
                                                    float* __restrict__ b1,
                                                    float* __restrict__ b2) {
  WaveAcc w = gemm_core<true>(P, DOUT, V, DOUT, DOUT,
                              blockIdx.x * BLK_M, blockIdx.y * BLK_N);
#pragma unroll
  for (int i = 0; i < 2; ++i)
#pragma unroll
    for (int j = 0; j < 2; ++j)
#pragma unroll
      for (int r = 0; r < 8; ++r) {
        const size_t o = (size_t)(w.rbase + i * 16 + r) * DC + (w.cbase + j * 16);
        const float g   = w.c[i][j][r];
        const float b1n = RHO * b1[o] + (1.f - RHO) * g * g;
        const float upd = LR * g / sqrtf(b1n + EPSR);
        const float b2n = MOM * b2[o] - upd;
        b1[o] = b1n;
        b2[o] = b2n;
        u[o] += MOM * b2n - upd;
      }
}

__global__ void k_zero(float* __restrict__ p, int n) {
  const int i = blockIdx.x * blockDim.x + threadIdx.x;
  if (i < n) p[i] = 0.f;
}

extern "C" void kernel_launch(void* const* d_in, const int* in_sizes, int n_in,
                              void* d_out, int out_size, void* d_ws, size_t ws_size,
                              hipStream_t stream) {
  const float* inputs = (const float*)d_in[0];   // [B, DIN]
  const float* W      = (const float*)d_in[1];   // [DOUT, DIN]
  const float* V      = (const float*)d_in[2];   // [DC, DOUT]

  float* x  = (float*)d_out;                     // [B, DOUT] final states

  float* ws = (float*)d_ws;
  const size_t NX = (size_t)B_DIM * DOUT;        // 4M
  const size_t NU = (size_t)B_DIM * DC;          // 512K
  float* a1 = ws;
  float* a2 = a1 + NX;
  float* u  = a2 + NX;
  float* b1 = u + NU;
  float* b2 = b1 + NU;
  float* EP = b2 + NU;                           // [B, DOUT]: E then P, per step
  float* Rm = EP + NX;                           // [B, DIN]

  // zero optimizer/state buffers (harness poisons ws/d_out)
  const int nzw = (int)(2 * NX + 3 * NU);
  k_zero<<<(nzw + 255) / 256, 256, 0, stream>>>(ws, nzw);
  k_zero<<<((int)NX + 255) / 256, 256, 0, stream>>>(x, (int)NX);

  const dim3 blk(256);
  const dim3 g1(B_DIM / BLK_M, DOUT / BLK_N);    // 16 x 32
  const dim3 g2(B_DIM / BLK_M, DIN / BLK_N);     // 16 x 16
  const dim3 g4(B_DIM / BLK_M, DC / BLK_N);      // 16 x 4

  for (int s = 0; s < NSTEP; ++s) {
    k1_uV_exp   <<<g1, blk, 0, stream>>>(u, V, EP);
    k2_xW_res   <<<g2, blk, 0, stream>>>(x, W, inputs, Rm);
    k3_gx_update<<<g1, blk, 0, stream>>>(Rm, W, x, a1, a2, EP);
    k4_gu_update<<<g4, blk, 0, stream>>>(EP, V, u, b1, b2);
  }
}